// EGraphSAGE_GraphAlign_22230750724513
// MI455X (gfx1250) — compile-verified
//
#include <hip/hip_runtime.h>

typedef float v2f __attribute__((ext_vector_type(2)));
typedef float v8f __attribute__((ext_vector_type(8)));
typedef float f4  __attribute__((ext_vector_type(4)));

#define NNODES 50000
#define NEDGES 800000

static_assert(NNODES % 16 == 0, "node tiles exact");
static_assert(NEDGES % 256 == 0, "edge blocks exact");

// ---------------------------------------------------------------------------
// Degree count: deg[d] += 1 per edge (f32 HW atomic; target is L2-resident)
// ---------------------------------------------------------------------------
__global__ __launch_bounds__(256) void count_kernel(const int* __restrict__ dst,
                                                    float* __restrict__ deg) {
  int e = blockIdx.x * 256 + threadIdx.x;   // grid covers E exactly
  unsafeAtomicAdd(&deg[dst[e]], 1.0f);
}

__global__ __launch_bounds__(256) void recip_kernel(float* __restrict__ deg, int n) {
  int i = blockIdx.x * 256 + threadIdx.x;
  if (i < n) deg[i] = 1.0f / fmaxf(deg[i], 1.0f);
}

// ---------------------------------------------------------------------------
// Fused message scatter: agg[d][0:64] += hn[src], agg[d][64:128] += he[e]
// 64 lanes per edge, 4 edges per 256-thread block.
// hn gather stays cached (L2-resident, 12.8MB); he is a 205MB stream -> NT.
// ---------------------------------------------------------------------------
__global__ __launch_bounds__(256) void aggregate_kernel(const float* __restrict__ hn,
                                                        const float* __restrict__ he,
                                                        const int* __restrict__ src,
                                                        const int* __restrict__ dst,
                                                        float* __restrict__ agg) {
  int j = threadIdx.x & 63;
  int e = blockIdx.x * 4 + (threadIdx.x >> 6);   // grid covers E exactly
  int s = src[e], d = dst[e];
  float ev = __builtin_nontemporal_load(&he[(size_t)e * 64 + j]);
  unsafeAtomicAdd(&agg[(size_t)d * 128 + j],      hn[(size_t)s * 64 + j]);
  unsafeAtomicAdd(&agg[(size_t)d * 128 + 64 + j], ev);
}

// ---------------------------------------------------------------------------
// Node MLP: out = relu([h | agg_n/deg | agg_e/deg] @ W + b)
//   X: [16 rows x 192] staged in LDS (scaled), W: [192 x 64] staged in LDS.
//   128 threads = 4 waves; each wave owns one 16x16 output tile.
//   K-loop: 48 x V_WMMA_F32_16X16X4_F32.
//   NOTE: h and out may alias (layer-1 in-place) -> no __restrict__ on them.
// ---------------------------------------------------------------------------
__global__ __launch_bounds__(128) void node_gemm_kernel(const float* h,
                                                        const float* __restrict__ agg,
                                                        const float* __restrict__ dinv,
                                                        const float* __restrict__ W,
                                                        const float* __restrict__ bias,
                                                        float* out) {
  __shared__ float Xs[16 * 196];   // pad 192 -> 196 to dodge bank conflicts
  __shared__ float Ws[192 * 65];   // pad 64 -> 65
  const int tid  = threadIdx.x;
  const int base = blockIdx.x * 16;          // grid covers N exactly

  for (int i = tid; i < 16 * 192; i += 128) {
    int m = i / 192, k = i - m * 192;
    int row = base + m;
    float v = (k < 64) ? h[(size_t)row * 64 + k]
                       : agg[(size_t)row * 128 + (k - 64)] * dinv[row];
    Xs[m * 196 + k] = v;
  }
  for (int i = tid; i < 192 * 64; i += 128) {
    int k = i >> 6, n = i & 63;
    Ws[k * 65 + n] = W[i];
  }
  __syncthreads();

  const int lane = tid & 31;
  const int wave = tid >> 5;
  const int m    = lane & 15;              // A row index == B col index per lane
  const int kb   = (lane >> 4) << 1;       // K sub-offset per ISA 16x4 layout
  const int ncol = wave * 16 + m;

  v8f c = {};
  for (int k = 0; k < 192; k += 4) {
    v2f a, b;
    a.x = Xs[m * 196 + k + kb];
    a.y = Xs[m * 196 + k + kb + 1];
    b.x = Ws[(k + kb)     * 65 + ncol];
    b.y = Ws[(k + kb + 1) * 65 + ncol];
    c = __builtin_amdgcn_wmma_f32_16x16x4_f32(false, a, false, b, (short)0, c, false, false);
  }

  const int rbase = base + ((lane >> 4) << 3);   // D layout: M = r + 8*(lane>=16)
  const float bb = bias[ncol];
  #pragma unroll
  for (int r = 0; r < 8; ++r) {
    out[(size_t)(rbase + r) * 64 + ncol] = fmaxf(c[r] + bb, 0.0f);
  }
}

// ---------------------------------------------------------------------------
// Per-node edge-MLP precompute: P[n] = [h[n] @ We_top | h[n] @ We_bot]
//   [50000,64] @ [64,128]; 256 threads = 8 waves = 8 col tiles, 16 rows/block.
// ---------------------------------------------------------------------------
__global__ __launch_bounds__(256) void p_gemm_kernel(const float* __restrict__ h,
                                                     const float* __restrict__ we,
                                                     float* __restrict__ P) {
  __shared__ float Hs [16 * 68];
  __shared__ float Ws2[64 * 130];
  const int tid  = threadIdx.x;
  const int base = blockIdx.x * 16;          // grid covers N exactly

  for (int i = tid; i < 16 * 64; i += 256) {
    int m = i >> 6, k = i & 63;
    Hs[m * 68 + k] = h[(size_t)(base + m) * 64 + k];
  }
  // W2[k][c] = we[k][c] (c<64, top half) or we[64+k][c-64] (bottom half)
  for (int i = tid; i < 64 * 128; i += 256) {
    int k = i >> 7, cc = i & 127;
    Ws2[k * 130 + cc] = (cc < 64) ? we[k * 64 + cc] : we[(64 + k) * 64 + (cc - 64)];
  }
  __syncthreads();

  const int lane = tid & 31;
  const int wave = tid >> 5;
  const int m    = lane & 15;
  const int kb   = (lane >> 4) << 1;
  const int ncol = wave * 16 + m;          // 0..127

  v8f c = {};
  for (int k = 0; k < 64; k += 4) {
    v2f a, b;
    a.x = Hs[m * 68 + k + kb];
    a.y = Hs[m * 68 + k + kb + 1];
    b.x = Ws2[(k + kb)     * 130 + ncol];
    b.y = Ws2[(k + kb + 1) * 130 + ncol];
    c = __builtin_amdgcn_wmma_f32_16x16x4_f32(false, a, false, b, (short)0, c, false, false);
  }

  const int rbase = base + ((lane >> 4) << 3);
  #pragma unroll
  for (int r = 0; r < 8; ++r) {
    P[(size_t)(rbase + r) * 128 + ncol] = c[r];
  }
}

// ---------------------------------------------------------------------------
// Edge update: he[e] = relu(P_top[src[e]] + P_bot[dst[e]] + b)
//   Pure bandwidth; P (25.6 MB) is L2-resident, output streams with NT stores
//   so it does not evict P. Indices staged through LDS (1 load per edge).
//   16 lanes/edge (float4 each), 16 edges per 256-thread block.
// ---------------------------------------------------------------------------
__global__ __launch_bounds__(256) void edge_kernel(const float* __restrict__ P,
                                                   const int* __restrict__ src,
                                                   const int* __restrict__ dst,
                                                   const float* __restrict__ bias,
                                                   float* __restrict__ he) {
  __shared__ int idx[32];                     // [0:16) = src, [16:32) = dst
  const int t  = threadIdx.x;
  const size_t e0 = (size_t)blockIdx.x * 16;  // grid covers E exactly
  if (t < 16)      idx[t]      = src[e0 + t];
  else if (t < 32) idx[t]      = dst[e0 + (t - 16)];
  __syncthreads();

  const int sub = t >> 4;                     // edge slot within block
  const int j   = (t & 15) << 2;              // float4 column
  const int s = idx[sub], d = idx[16 + sub];

  f4 pt = *(const f4*)(P + (size_t)s * 128 + j);
  f4 pb = *(const f4*)(P + (size_t)d * 128 + 64 + j);
  f4 bb = *(const f4*)(bias + j);
  f4 o;
  o.x = fmaxf(pt.x + pb.x + bb.x, 0.0f);
  o.y = fmaxf(pt.y + pb.y + bb.y, 0.0f);
  o.z = fmaxf(pt.z + pb.z + bb.z, 0.0f);
  o.w = fmaxf(pt.w + pb.w + bb.w, 0.0f);
  __builtin_nontemporal_store(o, (f4*)(he + (e0 + sub) * 64 + j));
}

// ---------------------------------------------------------------------------
extern "C" void kernel_launch(void* const* d_in, const int* in_sizes, int n_in,
                              void* d_out, int out_size, void* d_ws, size_t ws_size,
                              hipStream_t stream) {
  const float* nfeats = (const float*)d_in[0];
  const float* efeats = (const float*)d_in[1];
  const int*   src    = (const int*)d_in[2];
  const int*   dst    = (const int*)d_in[3];
  const float* wa0 = (const float*)d_in[4];
  const float* ba0 = (const float*)d_in[5];
  const float* we0 = (const float*)d_in[6];
  const float* be0 = (const float*)d_in[7];
  const float* wa1 = (const float*)d_in[8];
  const float* ba1 = (const float*)d_in[9];
  const float* we1 = (const float*)d_in[10];
  const float* be1 = (const float*)d_in[11];

  float* hn_out = (float*)d_out;                     // [N,64]
  float* he_out = hn_out + (size_t)NNODES * 64;      // [E,64]

  char*  ws  = (char*)d_ws;
  float* deg = (float*)ws;                                        // N floats (becomes 1/deg)
  float* agg = (float*)(ws + (((size_t)NNODES * 4 + 255) & ~(size_t)255)); // N*128
  float* P   = agg + (size_t)NNODES * 128;                        // N*128

  const int cnt_grid  = NEDGES / 256;
  const int rec_grid  = (NNODES + 255) / 256;
  const int agg_grid  = NEDGES / 4;
  const int node_grid = NNODES / 16;
  const int edge_grid = NEDGES / 16;

  // in-degree reciprocal (dst fixed across both layers)
  hipMemsetAsync(deg, 0, (size_t)NNODES * sizeof(float), stream);
  count_kernel<<<cnt_grid, 256, 0, stream>>>(dst, deg);
  recip_kernel<<<rec_grid, 256, 0, stream>>>(deg, NNODES);

  // ---- layer 0 ----
  hipMemsetAsync(agg, 0, (size_t)NNODES * 128 * sizeof(float), stream);
  aggregate_kernel<<<agg_grid, 256, 0, stream>>>(nfeats, efeats, src, dst, agg);
  node_gemm_kernel<<<node_grid, 128, 0, stream>>>(nfeats, agg, deg, wa0, ba0, hn_out);
  p_gemm_kernel<<<node_grid, 256, 0, stream>>>(hn_out, we0, P);
  edge_kernel<<<edge_grid, 256, 0, stream>>>(P, src, dst, be0, he_out);

  // ---- layer 1 (node GEMM runs in place on d_out; row-local so safe) ----
  hipMemsetAsync(agg, 0, (size_t)NNODES * 128 * sizeof(float), stream);
  aggregate_kernel<<<agg_grid, 256, 0, stream>>>(hn_out, he_out, src, dst, agg);
  node_gemm_kernel<<<node_grid, 128, 0, stream>>>(hn_out, agg, deg, wa1, ba1, hn_out);
  p_gemm_kernel<<<node_grid, 256, 0, stream>>>(hn_out, we1, P);
  edge_kernel<<<edge_grid, 256, 0, stream>>>(P, src, dst, be1, he_out);
}